// Operator2DNNBase_87746181857881
// MI455X (gfx1250) — compile-verified
//
#include <hip/hip_runtime.h>
#include <cstdint>
#include <cstddef>

// Problem constants (match reference: B=256, H=W=128, K=5)
#define HDIM 128
#define WDIM 128
#define BDIM 256
#define KW   5
#define NB   16           // batches per block (weight reuse factor)
#define SROWS 4           // output rows per block (strip height)
#define GROWS 8           // staged rows = SROWS + 4 halo
#define LROW 132          // 2 + 128 + 2 zero-padded LDS row

__global__ __launch_bounds__(512) void stencil5x5_async(
    const float* __restrict__ f,      // [B, H, W]
    const float* __restrict__ kern,   // [25, H*W]
    const float* __restrict__ dt,     // [B]
    float* __restrict__ out)          // [B, H, W]
{
    __shared__ float buf[2][GROWS][LROW];   // 8448 B, double-buffered window rows

    const int tid = threadIdx.x;            // 0..511
    const int jj  = tid & (WDIM - 1);       // column 0..127
    const int q   = tid >> 7;               // sub-row 0..3 (uniform per wave)
    const int si  = blockIdx.x & 31;        // row strip 0..31
    const int g   = blockIdx.x >> 5;        // batch group 0..15
    const int i0  = si * SROWS;             // first output row of strip
    const int ic  = i0 + q;                 // this thread's output row
    const int b0  = g * NB;
    const int pix = ic * WDIM + jj;

    // ---- one-time zeroing of padding: no div/mod, static pattern ----
    // Each thread handles LDS rows q and q+4 of each buffer.
    #pragma unroll
    for (int bufi = 0; bufi < 2; ++bufi) {
        #pragma unroll
        for (int half = 0; half < 2; ++half) {
            const int lr = q + 4 * half;          // 0..7
            const int r  = i0 + lr - 2;           // global row of this staged line
            const bool rowValid = (r >= 0) && (r < HDIM);
            if (!rowValid) {                      // fully-OOB row: zero everything
                buf[bufi][lr][jj] = 0.0f;
                if (jj < LROW - WDIM) buf[bufi][lr][WDIM + jj] = 0.0f;
            } else if (jj < 2) {                  // valid row: zero j-halo columns
                buf[bufi][lr][jj]       = 0.0f;
                buf[bufi][lr][130 + jj] = 0.0f;
            }
        }
    }

    // ---- per-pixel weights into registers; zero taps whose row is OOB ----
    float w[KW * KW];
    #pragma unroll
    for (int p = 0; p < KW * KW; ++p) {
        const int di = p / KW;
        const int r  = ic + di - 2;
        const float kv = kern[p * (HDIM * WDIM) + pix];   // coalesced across jj
        w[p] = (r >= 0 && r < HDIM) ? kv : 0.0f;          // OOB row -> tap contributes 0
    }

    const uint64_t gbase = (uint64_t)(uintptr_t)f;

    // Issue 2 async global->LDS row-segment loads for batch b into buffer bufi.
    // Every thread always issues exactly 2 (row clamped; OOB rows neutralized
    // by zeroed weights) so per-wave ASYNCcnt bookkeeping is uniform.
    auto issue = [&](int b, int bufi) {
        #pragma unroll
        for (int h = 0; h < 2; ++h) {
            const int lr = 2 * q + h;                         // 0..7
            int r  = i0 + lr - 2;
            int rc = r < 0 ? 0 : (r > HDIM - 1 ? HDIM - 1 : r);
            const uint64_t gaddr = gbase +
                (((uint64_t)b * (HDIM * WDIM)) + (uint64_t)rc * WDIM + (uint64_t)jj) * 4ull;
            const uint32_t laddr = (uint32_t)(uintptr_t)(&buf[bufi][lr][2 + jj]);
            asm volatile("global_load_async_to_lds_b32 %0, %1, off"
                         :: "v"(laddr), "v"(gaddr) : "memory");
        }
    };

    // ---- prologue: stage batch b0 into buffer 0 ----
    issue(b0, 0);

    // ---- main loop over the batch group, double-buffered ----
    for (int t = 0; t < NB; ++t) {
        if (t + 1 < NB) {
            issue(b0 + t + 1, (t + 1) & 1);
            // 2 loads in flight for the next buffer; async loads complete in
            // order, so cnt<=2 means buffer t's 2 loads have landed.
            asm volatile("s_wait_asynccnt 2" ::: "memory");
        } else {
            asm volatile("s_wait_asynccnt 0" ::: "memory");
        }
        __syncthreads();   // all waves' buffer-t rows visible to all threads

        const int bi = t & 1;
        float acc = 0.0f;
        #pragma unroll
        for (int di = 0; di < KW; ++di) {
            #pragma unroll
            for (int dj = 0; dj < KW; ++dj) {
                acc = fmaf(buf[bi][q + di][jj + dj], w[di * KW + dj], acc);
            }
        }
        const float fc  = buf[bi][q + 2][2 + jj];   // center tap == f[b,ic,jj]
        const float dtb = dt[b0 + t];               // uniform scalar load
        float res = fmaf(acc, dtb, fc);
        res = fmaxf(res, 0.0f);
        __builtin_nontemporal_store(res,
            &out[(size_t)(b0 + t) * (HDIM * WDIM) + pix]);   // streamed, never re-read

        __syncthreads();   // everyone done reading buf[t&1] before it is refilled
    }
}

extern "C" void kernel_launch(void* const* d_in, const int* in_sizes, int n_in,
                              void* d_out, int out_size, void* d_ws, size_t ws_size,
                              hipStream_t stream) {
    const float* f    = (const float*)d_in[0];   // [256,128,128]
    const float* kern = (const float*)d_in[1];   // [25, 16384]
    const float* dt   = (const float*)d_in[2];   // [256]
    float* out        = (float*)d_out;           // [256,128,128]

    (void)in_sizes; (void)n_in; (void)out_size; (void)d_ws; (void)ws_size;

    dim3 grid((HDIM / SROWS) * (BDIM / NB));     // 32 strips x 16 batch groups = 512 blocks
    dim3 block(512);                             // 16 wave32
    stencil5x5_async<<<grid, block, 0, stream>>>(f, kern, dt, out);
}